// TinyMambaClassifier_11613591569112
// MI455X (gfx1250) — compile-verified
//
#include <hip/hip_runtime.h>
#include <hip/hip_bf16.h>

#define D_MODEL 192
#define N_LAYERS 8
#define D_STATE 64
#define D_INNER 384
#define DT_RANK 12
#define NB 8
#define SEQ 1025                    // 1024 tokens + cls
#define MROWS (NB * SEQ)            // 8200
#define MTILES ((MROWS + 15) / 16)  // 513
#define MPAD (MTILES * 16)          // 8208
#define DBL_N (DT_RANK + 2 * D_STATE) // 140
#define DBL_LD 144
#define EPSC 1e-6f

#define KC 32                       // K-chunk staged in LDS
#define LSTR 36                     // padded LDS row stride (floats): conflict-free frags

typedef float v2f __attribute__((ext_vector_type(2)));
typedef float v4f __attribute__((ext_vector_type(4)));
typedef float v8f __attribute__((ext_vector_type(8)));

#define WMMA_F32(a, b, c) \
  __builtin_amdgcn_wmma_f32_16x16x4_f32(false, (a), false, (b), (short)0, (c), false, false)

// ---------------------------------------------------------------- embedding
__global__ void embed_k(const int* __restrict__ ids, const float* __restrict__ emb,
                        const float* __restrict__ cls, float* __restrict__ X) {
  int idx = blockIdx.x * blockDim.x + threadIdx.x;
  if (idx >= MROWS * D_MODEL) return;
  int c = idx % D_MODEL;
  int row = idx / D_MODEL;
  int t = row % SEQ, b = row / SEQ;
  float v = (t < 1024) ? emb[(size_t)ids[b * 1024 + t] * D_MODEL + c] : cls[c];
  X[idx] = v;
}

// ---------------------------------------------------------------- rmsnorm (wave per row)
__global__ void rmsnorm_k(const float* __restrict__ X, const float* __restrict__ w,
                          float* __restrict__ Out, int Mvalid) {
  int wave = threadIdx.x >> 5;
  int lane = threadIdx.x & 31;
  int row = blockIdx.x * 8 + wave;
  if (row >= Mvalid) return;
  const float* xr = X + (size_t)row * D_MODEL;
  float v[6];
  float ss = 0.f;
#pragma unroll
  for (int j = 0; j < 6; ++j) { v[j] = xr[lane + 32 * j]; ss += v[j] * v[j]; }
#pragma unroll
  for (int off = 16; off; off >>= 1) ss += __shfl_xor(ss, off, 32);
  float inv = rsqrtf(ss * (1.0f / D_MODEL) + EPSC);
  float* orow = Out + (size_t)row * D_MODEL;
#pragma unroll
  for (int j = 0; j < 6; ++j) orow[lane + 32 * j] = w[lane + 32 * j] * v[j] * inv;
}

// ---------------------------------------------------------------- WMMA f32 GEMM: C[M,N] = A[M,K] * W[N,K]^T
// Block = 4 waves -> 64x64 C tile. K staged through double-buffered LDS in
// chunks of 32: global_load_b128 (next chunk) -> 32x V_WMMA from ds_load_b64
// (current chunk) -> ds_store_b128 -> barrier. W fragments shared by all waves.
__global__ __launch_bounds__(128) void gemm_lds(
    const float* __restrict__ A, int lda,
    const float* __restrict__ W, int K, int Nvalid,
    float* __restrict__ C, int ldc, int Mvalid, int accum) {
  __shared__ __align__(16) float sA[2][64 * LSTR];
  __shared__ __align__(16) float sW[2][64 * LSTR];
  int tid = threadIdx.x;           // 0..127
  int wave = tid >> 5, lane = tid & 31;
  int half = lane >> 4, mn = lane & 15;
  int mbase = blockIdx.y * 64;
  int nbase = blockIdx.x * 64;

  // ---- staging map: thread -> 4 rows (i*16 + tid/8), quad kq = tid%8
  int r0 = tid >> 3;               // 0..15
  int kk = (tid & 7) * 4;          // 0,4,..,28 within chunk
  const float* Aq[4];
  const float* Wq[4];
#pragma unroll
  for (int i = 0; i < 4; ++i) {
    int r = i * 16 + r0;
    int gm = mbase + r; if (gm > MPAD - 1) gm = MPAD - 1;       // clamp (store-guarded)
    int gn = nbase + r; if (gn > Nvalid - 1) gn = Nvalid - 1;
    Aq[i] = A + (size_t)gm * lda;
    Wq[i] = W + (size_t)gn * K;
  }

  // ---- prologue: chunk 0 -> LDS buf 0 (W zero-filled past K)
  {
    v4f ra[4], rw[4];
#pragma unroll
    for (int i = 0; i < 4; ++i) {
      ra[i] = *(const v4f*)(Aq[i] + kk);                        // in-bounds (lda >= KC)
      v4f z = {};
      rw[i] = (kk < K) ? *(const v4f*)(Wq[i] + kk) : z;
    }
#pragma unroll
    for (int i = 0; i < 4; ++i) {
      int off = (i * 16 + r0) * LSTR + kk;
      *(v4f*)&sA[0][off] = ra[i];
      *(v4f*)&sW[0][off] = rw[i];
    }
  }
  __syncthreads();

  int nch = (K + KC - 1) / KC;
  v8f acc0 = {}, acc1 = {}, acc2 = {}, acc3 = {};
  v4f ra[4], rw[4];
  for (int c = 0; c < nch; ++c) {
    int cur = c & 1;
    bool more = (c + 1) < nch;                                  // uniform branch
    if (more) {                                                 // issue next-chunk global loads
      int kb = (c + 1) * KC + kk;                               // kb < K (K % KC == 0 when nch > 1)
#pragma unroll
      for (int i = 0; i < 4; ++i) {
        ra[i] = *(const v4f*)(Aq[i] + kb);
        rw[i] = *(const v4f*)(Wq[i] + kb);
      }
    }
    // ---- 8 k-steps x 4 WMMAs from LDS
    const float* LA  = &sA[cur][(wave * 16 + mn) * LSTR + 2 * half];
    const float* LB0 = &sW[cur][( 0 + mn) * LSTR + 2 * half];
    const float* LB1 = &sW[cur][(16 + mn) * LSTR + 2 * half];
    const float* LB2 = &sW[cur][(32 + mn) * LSTR + 2 * half];
    const float* LB3 = &sW[cur][(48 + mn) * LSTR + 2 * half];
#pragma unroll
    for (int ks = 0; ks < KC; ks += 4) {
      v2f a  = *(const v2f*)(LA  + ks);
      v2f b0 = *(const v2f*)(LB0 + ks);
      v2f b1 = *(const v2f*)(LB1 + ks);
      v2f b2 = *(const v2f*)(LB2 + ks);
      v2f b3 = *(const v2f*)(LB3 + ks);
      acc0 = WMMA_F32(a, b0, acc0);
      acc1 = WMMA_F32(a, b1, acc1);
      acc2 = WMMA_F32(a, b2, acc2);
      acc3 = WMMA_F32(a, b3, acc3);
    }
    if (more) {                                                 // park next chunk in other buffer
      int nxt = cur ^ 1;
#pragma unroll
      for (int i = 0; i < 4; ++i) {
        int off = (i * 16 + r0) * LSTR + kk;
        *(v4f*)&sA[nxt][off] = ra[i];
        *(v4f*)&sW[nxt][off] = rw[i];
      }
    }
    __syncthreads();
  }

  // ---- epilogue: guarded stores
  v8f accs[4] = {acc0, acc1, acc2, acc3};
#pragma unroll
  for (int j = 0; j < 4; ++j) {
    int col = nbase + j * 16 + mn;
    if (col < Nvalid) {
#pragma unroll
      for (int r = 0; r < 8; ++r) {
        int row = mbase + wave * 16 + half * 8 + r;             // C/D: VGPR r, lane-half -> M
        if (row < Mvalid) {
          size_t o = (size_t)row * ldc + col;
          if (accum) C[o] += accs[j][r]; else C[o] = accs[j][r];
        }
      }
    }
  }
}

// ---------------------------------------------------------------- depthwise causal conv3 + SiLU
__global__ void conv_silu_k(const float* __restrict__ XZ, const float* __restrict__ cw,
                            const float* __restrict__ cb, float* __restrict__ XI) {
  int idx = blockIdx.x * blockDim.x + threadIdx.x;
  if (idx >= MROWS * D_INNER) return;
  int c = idx % D_INNER;
  int row = idx / D_INNER;
  int t = row % SEQ, b = row / SEQ;
  float acc = cb[c];
#pragma unroll
  for (int j = 0; j < 3; ++j) {
    int tt = t - 2 + j;
    if (tt >= 0) acc += cw[c * 3 + j] * XZ[(size_t)(b * SEQ + tt) * 768 + c];
  }
  XI[idx] = acc / (1.f + __expf(-acc));
}

// ---------------------------------------------------------------- dt = softplus(dt + b)
__global__ void dt_softplus_k(float* __restrict__ DT, const float* __restrict__ dtb) {
  int idx = blockIdx.x * blockDim.x + threadIdx.x;
  if (idx >= MROWS * D_INNER) return;
  float x = DT[idx] + dtb[idx % D_INNER];
  DT[idx] = fmaxf(x, 0.f) + log1pf(__expf(-fabsf(x)));
}

// ---------------------------------------------------------------- selective scan
// block = 256 thr = 8 waves = 8 d-channels; 2 states/lane (64 states per wave).
// Double-buffered LDS for B_t/C_t: loads for t+1 overlap compute of t.
__global__ void scan_k(const float* __restrict__ DT, const float* __restrict__ XI,
                       const float* __restrict__ DBL, const float* __restrict__ Alog,
                       float* __restrict__ Y) {
  __shared__ float sB[2][D_STATE];
  __shared__ float sC[2][D_STATE];
  int b = blockIdx.x / 48;
  int dbase = (blockIdx.x % 48) * 8;
  int tid = threadIdx.x;
  int wave = tid >> 5;
  int lane = tid & 31;
  int d = dbase + wave;
  int n0 = lane * 2;
  float a0 = -__expf(Alog[(size_t)d * D_STATE + n0]);
  float a1 = -__expf(Alog[(size_t)d * D_STATE + n0 + 1]);
  float h0 = 0.f, h1 = 0.f;
  {
    size_t row0 = (size_t)b * SEQ;
    if (tid < D_STATE)            sB[0][tid] = DBL[row0 * DBL_LD + DT_RANK + tid];
    else if (tid < 2 * D_STATE)   sC[0][tid - D_STATE] =
        DBL[row0 * DBL_LD + DT_RANK + D_STATE + (tid - D_STATE)];
  }
  size_t row = (size_t)b * SEQ;
  float dt = DT[row * D_INNER + d];
  float u  = XI[row * D_INNER + d];
  __syncthreads();
  for (int t = 0; t < SEQ; ++t) {
    int p = t & 1;
    float dtn = 0.f, un = 0.f;
    if (t + 1 < SEQ) {                         // uniform branch
      size_t rn = row + 1;
      if (tid < D_STATE)          sB[p ^ 1][tid] = DBL[rn * DBL_LD + DT_RANK + tid];
      else if (tid < 2 * D_STATE) sC[p ^ 1][tid - D_STATE] =
          DBL[rn * DBL_LD + DT_RANK + D_STATE + (tid - D_STATE)];
      dtn = DT[rn * D_INNER + d];
      un  = XI[rn * D_INNER + d];
    }
    float dtu = dt * u;
    h0 = __expf(dt * a0) * h0 + dtu * sB[p][n0];
    h1 = __expf(dt * a1) * h1 + dtu * sB[p][n0 + 1];
    float s = h0 * sC[p][n0] + h1 * sC[p][n0 + 1];
#pragma unroll
    for (int off = 16; off; off >>= 1) s += __shfl_xor(s, off, 32);
    if (lane == 0) Y[row * D_INNER + d] = s;
    __syncthreads();                           // t reads of buf p done before t+1 overwrites
    dt = dtn; u = un; ++row;
  }
}

// ---------------------------------------------------------------- y = (y + xi*D) * silu(z)
__global__ void gate_k(float* __restrict__ Y, const float* __restrict__ XI,
                       const float* __restrict__ XZ, const float* __restrict__ Dp) {
  int idx = blockIdx.x * blockDim.x + threadIdx.x;
  if (idx >= MROWS * D_INNER) return;
  int d = idx % D_INNER;
  int row = idx / D_INNER;
  float z = XZ[(size_t)row * 768 + D_INNER + d];
  float y = Y[idx] + XI[idx] * Dp[d];
  Y[idx] = y * (z / (1.f + __expf(-z)));
}

// ---------------------------------------------------------------- mean-pool (first 1024) + head
__global__ void pool_head_k(const float* __restrict__ HN, const float* __restrict__ hw,
                            const float* __restrict__ hb, float* __restrict__ out) {
  __shared__ float sh[D_MODEL];
  int b = blockIdx.x;
  int dm = threadIdx.x;
  float s = 0.f;
  for (int t = 0; t < 1024; ++t) s += HN[(size_t)(b * SEQ + t) * D_MODEL + dm];
  sh[dm] = s * (1.f / 1024.f);
  __syncthreads();
  if (dm < 2) {
    float dot = hb[dm];
    for (int k = 0; k < D_MODEL; ++k) dot += sh[k] * hw[dm * D_MODEL + k];
    out[b * 2 + dm] = dot;
  }
}

// ================================================================ host
extern "C" void kernel_launch(void* const* d_in, const int* in_sizes, int n_in,
                              void* d_out, int out_size, void* d_ws, size_t ws_size,
                              hipStream_t stream) {
  (void)in_sizes; (void)n_in; (void)out_size; (void)ws_size;
  const int*   ids    = (const int*)  d_in[0];
  const float* embed  = (const float*)d_in[2];
  const float* cls    = (const float*)d_in[3];
  const float* norm_w = (const float*)d_in[4];
  const float* in_w   = (const float*)d_in[5];
  const float* conv_w = (const float*)d_in[6];
  const float* conv_b = (const float*)d_in[7];
  const float* xp_w   = (const float*)d_in[8];
  const float* dt_w   = (const float*)d_in[9];
  const float* dt_b   = (const float*)d_in[10];
  const float* A_log  = (const float*)d_in[11];
  const float* D_p    = (const float*)d_in[12];
  const float* out_w  = (const float*)d_in[13];
  const float* fn_w   = (const float*)d_in[14];
  const float* hd_w   = (const float*)d_in[15];
  const float* hd_b   = (const float*)d_in[16];

  float* ws  = (float*)d_ws;
  float* X   = ws;                                  // MPAD*192 residual stream
  float* HN  = X   + (size_t)MPAD * D_MODEL;        // MPAD*192 rmsnorm out
  float* XZ  = HN  + (size_t)MPAD * D_MODEL;        // MPAD*768 in_proj out
  float* XI  = XZ  + (size_t)MPAD * 768;            // MPAD*384 conv+silu
  float* DBLb= XI  + (size_t)MPAD * D_INNER;        // MPAD*144 x_proj out (dt_r|B|C)
  float* DTb = DBLb+ (size_t)MPAD * DBL_LD;         // MPAD*384 dt
  float* Yb  = DTb + (size_t)MPAD * D_INNER;        // MPAD*384 scan out

  dim3 gw(128);
  int gy = (MTILES + 3) / 4;           // 64 M-rows per block
  int eg = (MROWS * D_INNER + 255) / 256;

  int gx_in  = 768 / 64;               // 12
  int gx_dbl = (DBL_N + 63) / 64;      // 3 (cols >= 140 clamped)
  int gx_dt  = D_INNER / 64;           // 6
  int gx_out = D_MODEL / 64;           // 3

  embed_k<<<(MROWS * D_MODEL + 255) / 256, 256, 0, stream>>>(ids, embed, cls, X);
  for (int l = 0; l < N_LAYERS; ++l) {
    rmsnorm_k<<<(MROWS + 7) / 8, 256, 0, stream>>>(X, norm_w + l * D_MODEL, HN, MROWS);
    gemm_lds<<<dim3(gx_in, gy), gw, 0, stream>>>(HN, D_MODEL,
        in_w + (size_t)l * 768 * D_MODEL, D_MODEL, 768, XZ, 768, MROWS, 0);
    conv_silu_k<<<eg, 256, 0, stream>>>(XZ, conv_w + l * D_INNER * 3,
        conv_b + l * D_INNER, XI);
    gemm_lds<<<dim3(gx_dbl, gy), gw, 0, stream>>>(XI, D_INNER,
        xp_w + (size_t)l * DBL_N * D_INNER, D_INNER, DBL_N, DBLb, DBL_LD, MROWS, 0);
    gemm_lds<<<dim3(gx_dt, gy), gw, 0, stream>>>(DBLb, DBL_LD,
        dt_w + (size_t)l * D_INNER * DT_RANK, DT_RANK, D_INNER, DTb, D_INNER, MROWS, 0);
    dt_softplus_k<<<eg, 256, 0, stream>>>(DTb, dt_b + l * D_INNER);
    scan_k<<<NB * 48, 256, 0, stream>>>(DTb, XI, DBLb,
        A_log + (size_t)l * D_INNER * D_STATE, Yb);
    gate_k<<<eg, 256, 0, stream>>>(Yb, XI, XZ, D_p + l * D_INNER);
    gemm_lds<<<dim3(gx_out, gy), gw, 0, stream>>>(Yb, D_INNER,
        out_w + (size_t)l * D_MODEL * D_INNER, D_INNER, D_MODEL, X, D_MODEL, MROWS, 1);
  }
  rmsnorm_k<<<(MROWS + 7) / 8, 256, 0, stream>>>(X, fn_w, HN, MROWS);
  pool_head_k<<<NB, D_MODEL, 0, stream>>>(HN, hd_w, hd_b, (float*)d_out);
}